// HarmonicOscillator_35734127903004
// MI455X (gfx1250) — compile-verified
//
#include <hip/hip_runtime.h>
#include <math.h>

// ---------------- problem constants ----------------
#define NB      4
#define CCH     256       // input channels
#define LF      256       // frames
#define LW      245760    // output samples per batch (LF * 960)
#define NH      32        // harmonics
#define CHUNK   1024      // samples per scan chunk
#define NCHUNK  240       // chunks per batch (LW / CHUNK)
#define TWO_PI_F 6.283185307179586f
#define INV_SR  (1.0f / 48000.0f)
#define SCALE_F ((float)LF / (float)LW)   // 1/960

typedef __attribute__((ext_vector_type(2))) float v2f;
typedef __attribute__((ext_vector_type(8))) float v8f;

// ---------------------------------------------------------------------------
// Kernel 1: mag = exp(min(W_mag @ x + b, 4)) via fp32 WMMA 16x16x4.
// Tiles: M(32 harmonics)=2 tiles, N(4*256 columns)=64 tiles, K=256 (64 steps).
// One wave per output tile; 128 waves = 32 blocks x 128 threads.
// Output layout mag_s[(n*LF + l)*NH + h] (frame-major for synth kernel).
// ---------------------------------------------------------------------------
__global__ void mag_gemm_wmma(const float* __restrict__ x,
                              const float* __restrict__ w_mag,
                              const float* __restrict__ b_mag,
                              float* __restrict__ mag_s) {
    const int wave = (blockIdx.x * blockDim.x + threadIdx.x) >> 5;  // 0..127
    const int lane = threadIdx.x & 31;
    const int mt   = wave >> 6;          // 0..1   (harmonic tile)
    const int pt   = wave & 63;          // 0..63  (column tile)
    const int n    = pt >> 4;            // batch
    const int l0   = (pt & 15) << 4;     // frame base
    const int half = lane >> 4;          // 0/1
    const int mn   = lane & 15;          // A row (M) and B col (N)

    const float* wrow = w_mag + (mt * 16 + mn) * CCH;       // A: row m
    const float* xcol = x + n * (CCH * LF) + (l0 + mn);     // B: column n

    v8f acc = {};
    for (int c = 0; c < CCH; c += 4) {
        const int kA = c + half * 2;
        v2f a, b;
        // A (16x4): lanes<16 hold K={0,1}, lanes>=16 hold K={2,3}
        a.x = wrow[kA];
        a.y = wrow[kA + 1];
        // B (4x16): same half-wave K split, N = lane&15
        b.x = xcol[kA * LF];
        b.y = xcol[(kA + 1) * LF];
        acc = __builtin_amdgcn_wmma_f32_16x16x4_f32(
            /*neg_a=*/false, a, /*neg_b=*/false, b,
            /*c_mod=*/(short)0, acc, /*reuse_a=*/false, /*reuse_b=*/false);
    }

    const int l = l0 + mn;
#pragma unroll
    for (int v = 0; v < 8; ++v) {
        const int h = mt * 16 + half * 8 + v;   // C/D: M = v + 8*half
        float val = acc[v] + b_mag[h];
        val = expf(fminf(val, 4.0f));
        mag_s[(n * LF + l) * NH + h] = val;
    }
}

// ---------------------------------------------------------------------------
// Kernel 2: f0[n,l] = clip(exp2(atan(a/(b+1e-4))/(2pi)*s1)*s2, 20, 2000)
// where [a;b] = W_f0 @ x + b_f0.  1024 threads, one per (n,l).
// ---------------------------------------------------------------------------
__global__ void f0_kernel(const float* __restrict__ x,
                          const float* __restrict__ w_f0,
                          const float* __restrict__ b_f0,
                          const float* __restrict__ s1,
                          const float* __restrict__ s2,
                          float* __restrict__ f0_s) {
    const int idx = blockIdx.x * blockDim.x + threadIdx.x;  // 0..1023
    const int n = idx >> 8, l = idx & 255;
    const float* xp = x + n * (CCH * LF) + l;
    float acc0 = 0.0f, acc1 = 0.0f;
    for (int c = 0; c < CCH; ++c) {
        const float xv = xp[c * LF];
        acc0 = fmaf(w_f0[c], xv, acc0);
        acc1 = fmaf(w_f0[CCH + c], xv, acc1);
    }
    const float a = acc0 + b_f0[0];
    const float b = acc1 + b_f0[1];
    float f0 = atanf(a / (b + 1e-4f)) * (1.0f / TWO_PI_F);
    f0 = exp2f(f0 * s1[0]) * s2[0];
    f0 = fminf(fmaxf(f0, 20.0f), 2000.0f);
    f0_s[idx] = f0;
}

// linear interp of f0 frames at output sample i (align_corners=False)
__device__ __forceinline__ float f0_interp(const float* __restrict__ f0n, int i) {
    float src = ((float)i + 0.5f) * SCALE_F - 0.5f;
    src = fminf(fmaxf(src, 0.0f), (float)(LF - 1));
    const int i0 = (int)floorf(src);
    const int i1 = min(i0 + 1, LF - 1);
    const float w = src - (float)i0;
    return f0n[i0] * (1.0f - w) + f0n[i1] * w;
}

// ---------------------------------------------------------------------------
// Kernel 3: per-chunk sums of v_i = interp(f0)/SR.  960 blocks x 256 threads,
// 4 samples/thread, LDS tree reduction.
// ---------------------------------------------------------------------------
__global__ void chunk_sum_kernel(const float* __restrict__ f0_s,
                                 float* __restrict__ chunkSum) {
    const int b = blockIdx.x;                 // 0..959
    const int n = b / NCHUNK;
    const int base = (b % NCHUNK) * CHUNK;
    const float* f0n = f0_s + n * LF;
    const int t = threadIdx.x;
    float s = 0.0f;
#pragma unroll
    for (int j = 0; j < 4; ++j)
        s += f0_interp(f0n, base + t * 4 + j) * INV_SR;

    __shared__ float red[256];
    red[t] = s;
    __syncthreads();
    for (int off = 128; off > 0; off >>= 1) {
        if (t < off) red[t] += red[t + off];
        __syncthreads();
    }
    if (t == 0) chunkSum[b] = red[0];
}

// ---------------------------------------------------------------------------
// Kernel 4: exclusive fp64 scan of the 240 chunk sums per batch (tiny, serial).
// ---------------------------------------------------------------------------
__global__ void chunk_scan_kernel(const float* __restrict__ chunkSum,
                                  double* __restrict__ chunkOff) {
    const int n = threadIdx.x;
    if (n < NB) {
        double acc = 0.0;
        for (int k = 0; k < NCHUNK; ++k) {
            chunkOff[n * NCHUNK + k] = acc;
            acc += (double)chunkSum[n * NCHUNK + k];
        }
    }
}

// ---------------------------------------------------------------------------
// Kernel 5: synthesis.  Per 1024-sample chunk: in-block inclusive scan of v_i,
// T = chunkOff + prefix; theta = 2*pi*frac(T) (frac in fp64 keeps phase sane);
// wave = mean_h sin((h+1)*theta) * magI_h via Chebyshev recurrence.
// mag for the <=3 frames touched by a chunk is staged in LDS.
// ---------------------------------------------------------------------------
__global__ void synth_kernel(const float* __restrict__ f0_s,
                             const float* __restrict__ mag_s,
                             const double* __restrict__ chunkOff,
                             float* __restrict__ out) {
    const int b = blockIdx.x;                  // 0..959
    const int n = b / NCHUNK;
    const int ch = b % NCHUNK;
    const int base = ch * CHUNK;
    const int t = threadIdx.x;
    const float* f0n = f0_s + n * LF;

    // first frame this chunk can touch
    float src0 = ((float)base + 0.5f) * SCALE_F - 0.5f;
    src0 = fminf(fmaxf(src0, 0.0f), (float)(LF - 1));
    const int lmin = (int)floorf(src0);

    __shared__ float smag[3 * NH];
    if (t < 3 * NH) {
        const int j = t >> 5, h = t & 31;
        const int fr = min(lmin + j, LF - 1);
        smag[t] = mag_s[(n * LF + fr) * NH + h];
    }

    // per-thread 4-sample serial prefix
    float p[4], wts[4];
    int li0[4], li1[4];
    float run = 0.0f;
#pragma unroll
    for (int j = 0; j < 4; ++j) {
        const int i = base + t * 4 + j;
        float src = ((float)i + 0.5f) * SCALE_F - 0.5f;
        src = fminf(fmaxf(src, 0.0f), (float)(LF - 1));
        const int i0 = (int)floorf(src);
        const int i1 = min(i0 + 1, LF - 1);
        const float w = src - (float)i0;
        const float v = (f0n[i0] * (1.0f - w) + f0n[i1] * w) * INV_SR;
        run += v;
        p[j] = run;
        wts[j] = w; li0[j] = i0; li1[j] = i1;
    }

    // block-wide inclusive scan of per-thread sums (Hillis-Steele in LDS)
    __shared__ float sc[256];
    sc[t] = run;
    __syncthreads();
    for (int off = 1; off < 256; off <<= 1) {
        const float other = (t >= off) ? sc[t - off] : 0.0f;
        __syncthreads();
        sc[t] += other;
        __syncthreads();
    }
    const float excl = sc[t] - run;
    const double offd = chunkOff[n * NCHUNK + ch];

#pragma unroll
    for (int j = 0; j < 4; ++j) {
        const double td = offd + (double)(excl + p[j]);
        const float u = (float)(td - floor(td));       // frac(T) in [0,1)
        const float theta = u * TWO_PI_F;
        const float s1v = sinf(theta);
        const float c1v = cosf(theta);
        const float twoc = 2.0f * c1v;

        const float* m0p = smag + (li0[j] - lmin) * NH;
        const float* m1p = smag + (li1[j] - lmin) * NH;
        const float w = wts[j];

        float sk = s1v, skm1 = 0.0f, acc = 0.0f;
#pragma unroll 8
        for (int h = 0; h < NH; ++h) {
            const float m0 = m0p[h], m1 = m1p[h];
            const float mg = fmaf(w, m1 - m0, m0);     // interp mag
            acc = fmaf(sk, mg, acc);                   // += sin(k*theta)*mag
            const float sn = fmaf(twoc, sk, -skm1);    // sin((k+1)theta)
            skm1 = sk; sk = sn;
        }
        out[n * LW + base + t * 4 + j] = acc * (1.0f / 32.0f);
    }
}

// ---------------------------------------------------------------------------
extern "C" void kernel_launch(void* const* d_in, const int* in_sizes, int n_in,
                              void* d_out, int out_size, void* d_ws, size_t ws_size,
                              hipStream_t stream) {
    const float* x     = (const float*)d_in[0];
    const float* w_f0  = (const float*)d_in[1];
    const float* b_f0  = (const float*)d_in[2];
    const float* s1    = (const float*)d_in[3];
    const float* s2    = (const float*)d_in[4];
    const float* w_mag = (const float*)d_in[5];
    const float* b_mag = (const float*)d_in[6];
    float* out = (float*)d_out;

    // workspace layout (all offsets 8-byte aligned)
    char* ws = (char*)d_ws;
    float*  f0_s     = (float*)(ws);                       // 1024 f32
    float*  mag_s    = (float*)(ws + 4096);                // 32768 f32
    float*  chunkSum = (float*)(ws + 4096 + 131072);       // 960 f32
    double* chunkOff = (double*)(ws + 4096 + 131072 + 4096); // 960 f64 (off 139264, 8B aligned)

    hipLaunchKernelGGL(mag_gemm_wmma, dim3(32), dim3(128), 0, stream,
                       x, w_mag, b_mag, mag_s);
    hipLaunchKernelGGL(f0_kernel, dim3(4), dim3(256), 0, stream,
                       x, w_f0, b_f0, s1, s2, f0_s);
    hipLaunchKernelGGL(chunk_sum_kernel, dim3(960), dim3(256), 0, stream,
                       f0_s, chunkSum);
    hipLaunchKernelGGL(chunk_scan_kernel, dim3(1), dim3(32), 0, stream,
                       chunkSum, chunkOff);
    hipLaunchKernelGGL(synth_kernel, dim3(960), dim3(256), 0, stream,
                       f0_s, mag_s, chunkOff, out);
}